// ColorGANGenerator_74715251081431
// MI455X (gfx1250) — compile-verified
//
#include <hip/hip_runtime.h>
#include <math.h>

// ColorGAN LUT-affine-tanh:  out = tanh(weight[idx] * img + bias[idx])
// img: [16,3,512,512] f32, LUT tables: [256^3, 3] f32.
// Memory-bound; dominated by random 12B gathers into 402MB of LUT.
// Strategy: async global->LDS double-buffered streaming of img tiles
// (ASYNCcnt path, NT temporal hint so the stream doesn't evict LUT lines
// from the 192MB L2), default-temporal LUT gathers, hardware v_tanh_f32,
// non-temporal b128 stores for write-once output.

typedef __attribute__((ext_vector_type(4))) float floatx4;

namespace {
constexpr int kHW      = 512 * 512;   // plane size, 2^18
constexpr int kTilePix = 1024;        // pixels per tile per block (256 thr * 4)
}

static __device__ __forceinline__ float fast_tanh(float x) {
#if __has_builtin(__builtin_amdgcn_tanhf)
    return __builtin_amdgcn_tanhf(x);           // v_tanh_f32, hazards by compiler
#else
    float y;
    // TRANS op: per ISA, 1 delay before result use / source overwrite.
    asm("v_tanh_f32 %0, %1\n\t"
        "s_delay_alu instid0(TRANS32_DEP_1)"
        : "=v"(y) : "v"(x));
    return y;
#endif
}

__global__ __launch_bounds__(256) void ColorGANGenerator_74715251081431_kernel(
    const float* __restrict__ img,
    const float* __restrict__ weight,
    const float* __restrict__ bias,
    float* __restrict__ out,
    int num_tiles)
{
    // Double-buffered tile: [stage][channel][pixel] -> 24 KB LDS
    __shared__ float lds[2][3][kTilePix];

    const int tid   = threadIdx.x;
    // Each wave (32 lanes) owns a 128-pixel slice; each lane owns 4 pixels.
    const int chunk = ((tid >> 5) << 7) + ((tid & 31) << 2);

    const int t0 = blockIdx.x;
    if (t0 >= num_tiles) return;
    const int stride = gridDim.x;

    int s = 0;

    // Issue async copy (global -> LDS) of one tile's 3 channel slices.
    // Per wave: 3 x (32 lanes x b128) = 3 x 512B. Tracked by ASYNCcnt.
    // NT hint: img is read exactly once -> keep L2 for the LUT gathers.
    auto issue = [&](int tile, int stage) {
        const int P   = tile * kTilePix;
        const int b   = P >> 18;                 // / kHW
        const int off = (P & (kHW - 1)) + chunk;
        const float* base = img + (size_t)(b * 3) * kHW + off;
#pragma unroll
        for (int c = 0; c < 3; ++c) {
            unsigned     lad = (unsigned)(size_t)(&lds[stage][c][chunk]);
            const float* gad = base + (size_t)c * kHW;
            asm volatile("global_load_async_to_lds_b128 %0, %1, off th:TH_LOAD_NT"
                         :: "v"(lad), "v"(gad)
                         : "memory");
        }
    };

    issue(t0, 0);

    for (int t = t0; t < num_tiles; t += stride) {
        const int  tn       = t + stride;
        const bool has_next = tn < num_tiles;
        if (has_next) {
            issue(tn, s ^ 1);                            // prefetch next tile
            asm volatile("s_wait_asynccnt 0x3" ::: "memory");  // current tile done
        } else {
            asm volatile("s_wait_asynccnt 0x0" ::: "memory");
        }

        const int P   = t * kTilePix;
        const int b   = P >> 18;
        const int off = (P & (kHW - 1)) + chunk;

        const floatx4 rv = *reinterpret_cast<const floatx4*>(&lds[s][0][chunk]);
        const floatx4 gv = *reinterpret_cast<const floatx4*>(&lds[s][1][chunk]);
        const floatx4 bv = *reinterpret_cast<const floatx4*>(&lds[s][2][chunk]);

        float o0[4], o1[4], o2[4];
#pragma unroll
        for (int k = 0; k < 4; ++k) {
            const float r  = rv[k];
            const float g  = gv[k];
            const float bl = bv[k];
            // Match reference numerics: f32 quantize, f32 combine, trunc->i32.
            const float q0 = (r  + 1.0f) * 127.5f;
            const float q1 = (g  + 1.0f) * 127.5f;
            const float q2 = (bl + 1.0f) * 127.5f;
            const int   idx = (int)(q0 * 65536.0f + q1 * 256.0f + q2);
            // 12B rows -> global_load_b96 gathers; default temporal hint so
            // the 192MB L2 keeps as much of the 402MB LUT set as possible.
            const float* wr = weight + (size_t)idx * 3;
            const float* br = bias   + (size_t)idx * 3;
            o0[k] = fast_tanh(fmaf(wr[0], r,  br[0]));
            o1[k] = fast_tanh(fmaf(wr[1], g,  br[1]));
            o2[k] = fast_tanh(fmaf(wr[2], bl, br[2]));
        }

        // Write-once output: non-temporal b128 stores (don't pollute L2).
        float* obase = out + (size_t)(b * 3) * kHW + off;
        __builtin_nontemporal_store((floatx4){o0[0], o0[1], o0[2], o0[3]},
                                    reinterpret_cast<floatx4*>(obase));
        __builtin_nontemporal_store((floatx4){o1[0], o1[1], o1[2], o1[3]},
                                    reinterpret_cast<floatx4*>(obase + kHW));
        __builtin_nontemporal_store((floatx4){o2[0], o2[1], o2[2], o2[3]},
                                    reinterpret_cast<floatx4*>(obase + 2 * kHW));
        s ^= 1;
    }
}

extern "C" void kernel_launch(void* const* d_in, const int* in_sizes, int n_in,
                              void* d_out, int out_size, void* d_ws, size_t ws_size,
                              hipStream_t stream) {
    (void)n_in; (void)out_size; (void)d_ws; (void)ws_size;
    const float* img    = (const float*)d_in[0];
    const float* weight = (const float*)d_in[1];
    const float* bias   = (const float*)d_in[2];
    float*       out    = (float*)d_out;

    const int npix      = in_sizes[0] / 3;        // 16*512*512 = 4,194,304
    const int num_tiles = npix / kTilePix;        // 4096

    int blocks = num_tiles < 1024 ? num_tiles : 1024;  // 4 tiles/block, 3 overlapped
    ColorGANGenerator_74715251081431_kernel<<<blocks, 256, 0, stream>>>(
        img, weight, bias, out, num_tiles);
}